// AttentiveGraphPooling_61959198212372
// MI455X (gfx1250) — compile-verified
//
#include <hip/hip_runtime.h>
#include <hip/hip_bf16.h>
#include <math.h>

#define N_   256
#define NV_  2048
#define NE_  64
#define NH_  4

typedef __attribute__((ext_vector_type(2))) float v2f;
typedef __attribute__((ext_vector_type(8))) float v8f;

// ---------------------------------------------------------------------------
// Kernel 1: qh[h*64+e] = Wc[h*64+e, :] . query + bc   (256 rows, K=64)
// ---------------------------------------------------------------------------
__global__ __launch_bounds__(256) void qh_kernel(
    const float* __restrict__ Wc, const float* __restrict__ bc,
    const float* __restrict__ query, float* __restrict__ qh)
{
    __shared__ float s_q[NE_];
    const int t = threadIdx.x;
    if (t < NE_) s_q[t] = query[t];
    __syncthreads();
    float acc = bc[t];
    const float* w = Wc + t * NE_;
    #pragma unroll 8
    for (int e = 0; e < NE_; ++e) acc = fmaf(w[e], s_q[e], acc);
    qh[t] = acc;
}

// ---------------------------------------------------------------------------
// Kernel 2: per-graph attention pooling.
//   logits[v,h] = sum_e tanh(vert[v,e] + qh[h,e]) * wl[e] + bl
//   masked softmax over v (per h), then heads[h,e] = sum_v p*mask*vert
//   leaky_relu(0.01) applied; heads written to workspace (N x 256).
// One block (256 threads, 8 waves) per graph n.
// ---------------------------------------------------------------------------
__global__ __launch_bounds__(256) void pool_kernel(
    const float* __restrict__ vertices, const float* __restrict__ mask,
    const float* __restrict__ qh, const float* __restrict__ wl,
    const float* __restrict__ bl, const float* __restrict__ temp,
    float* __restrict__ heads)
{
    __shared__ float s_qhT[NE_][NH_];   // qh transposed: [e][h], rows are float4
    __shared__ float s_wl[NE_];
    __shared__ float s_p[NV_][NH_];     // logits -> masked exp weights (32 KB)
    __shared__ float s_red[256];
    __shared__ float s_max[NH_];
    __shared__ float s_inv[NH_];

    const int n = blockIdx.x;
    const int t = threadIdx.x;
    const float blv  = bl[0];
    const float invT = 1.0f / temp[0];

    if (t < NE_) s_wl[t] = wl[t];
    {   // qh[h*64+e] -> s_qhT[e][h]
        const int h = t >> 6, e = t & 63;
        s_qhT[e][h] = qh[t];
    }
    __syncthreads();

    const float* vbase = vertices + (size_t)n * NV_ * NE_;
    const float* mbase = mask + (size_t)n * NV_;
    const float4* qhT4 = (const float4*)&s_qhT[0][0];

    // ---- Phase A: masked logits into LDS -------------------------------
    for (int vb = 0; vb < NV_; vb += 256) {
        const int v = vb + t;
        const float m = mbase[v];
        const float* vr = vbase + (size_t)v * NE_;
        float a0 = blv, a1 = blv, a2 = blv, a3 = blv;
        #pragma unroll 4
        for (int e = 0; e < NE_; e += 4) {
            float4 x = *(const float4*)(vr + e);
            float xs[4] = { x.x, x.y, x.z, x.w };
            #pragma unroll
            for (int j = 0; j < 4; ++j) {
                const float4 q = qhT4[e + j];
                const float w  = s_wl[e + j];
                a0 = fmaf(tanhf(xs[j] + q.x), w, a0);
                a1 = fmaf(tanhf(xs[j] + q.y), w, a1);
                a2 = fmaf(tanhf(xs[j] + q.z), w, a2);
                a3 = fmaf(tanhf(xs[j] + q.w), w, a3);
            }
        }
        const float neg = -99999.0f * (1.0f - m);
        s_p[v][0] = a0 * invT * m + neg;
        s_p[v][1] = a1 * invT * m + neg;
        s_p[v][2] = a2 * invT * m + neg;
        s_p[v][3] = a3 * invT * m + neg;
    }
    __syncthreads();

    // ---- softmax reductions: t = idx*4 + h, idx in [0,64) ---------------
    const int h   = t & 3;
    const int idx = t >> 2;

    float lmax = -3.402823466e38f;
    for (int v = idx; v < NV_; v += 64) lmax = fmaxf(lmax, s_p[v][h]);
    s_red[t] = lmax;
    __syncthreads();
    for (int s = 128; s >= 4; s >>= 1) {
        if (t < s) s_red[t] = fmaxf(s_red[t], s_red[t + s]);
        __syncthreads();
    }
    if (t < NH_) s_max[t] = s_red[t];
    __syncthreads();

    const float mh = s_max[h];
    float lsum = 0.0f;
    for (int v = idx; v < NV_; v += 64) {
        const float p = __expf(s_p[v][h] - mh);
        lsum += p;                      // softmax denominator (unmasked)
        s_p[v][h] = p * mbase[v];       // masked numerator weight for pooling
    }
    s_red[t] = lsum;
    __syncthreads();
    for (int s = 128; s >= 4; s >>= 1) {
        if (t < s) s_red[t] += s_red[t + s];
        __syncthreads();
    }
    if (t < NH_) s_inv[t] = 1.0f / s_red[t];
    __syncthreads();

    // ---- Phase B: heads[h,e] = inv_sum * sum_v p[v,h]*vert[v,e] ---------
    const int e  = t & 63;
    const int hh = t >> 6;
    const float inv = s_inv[hh];
    float acc = 0.0f;
    for (int v = 0; v < NV_; v += 4) {
        acc = fmaf(s_p[v + 0][hh], vbase[(size_t)(v + 0) * NE_ + e], acc);
        acc = fmaf(s_p[v + 1][hh], vbase[(size_t)(v + 1) * NE_ + e], acc);
        acc = fmaf(s_p[v + 2][hh], vbase[(size_t)(v + 2) * NE_ + e], acc);
        acc = fmaf(s_p[v + 3][hh], vbase[(size_t)(v + 3) * NE_ + e], acc);
    }
    float hv = acc * inv;
    hv = hv > 0.0f ? hv : 0.01f * hv;   // leaky_relu
    heads[(size_t)n * (NH_ * NE_) + t] = hv;
}

// ---------------------------------------------------------------------------
// Kernel 3: out(256x64) = heads(256x256) @ Wr^T(256x64) + br
// WMMA f32 16x16x4, one 16x16 tile per wave. 64 waves total (8 blocks x 8).
// A layout (16x4 f32): lanes 0-15 hold rows M=0..15, vgpr0=K, vgpr1=K+1;
// lanes 16-31 hold K+2, K+3. B mirrored. D = documented 16x16 f32 striping.
// ---------------------------------------------------------------------------
__global__ __launch_bounds__(256) void proj_kernel(
    const float* __restrict__ heads, const float* __restrict__ Wr,
    const float* __restrict__ br, float* __restrict__ out)
{
    const int wave = threadIdx.x >> 5;
    const int lane = threadIdx.x & 31;
    const int tile = blockIdx.x * 8 + wave;   // 64 tiles: 16 (M) x 4 (N)
    const int m0 = (tile >> 2) << 4;
    const int n0 = (tile & 3) << 4;
    const int lm   = lane & 15;
    const int half = lane >> 4;               // selects K pair

    const float* arow = heads + (size_t)(m0 + lm) * (NH_ * NE_);
    const float* brow = Wr    + (size_t)(n0 + lm) * (NH_ * NE_);  // Wr[n][k]

    v8f c = {};
    #pragma unroll 4
    for (int k = 0; k < NH_ * NE_; k += 4) {
        const int kk = k + half * 2;
        v2f a; a.x = arow[kk]; a.y = arow[kk + 1];
        v2f b; b.x = brow[kk]; b.y = brow[kk + 1];
        c = __builtin_amdgcn_wmma_f32_16x16x4_f32(
                false, a, false, b, (short)0, c, false, false);
    }

    const float bias = br[n0 + lm];
    #pragma unroll
    for (int i = 0; i < 8; ++i) {
        const int row = m0 + i + half * 8;
        out[(size_t)row * NE_ + n0 + lm] = c[i] + bias;
    }
}

// ---------------------------------------------------------------------------
extern "C" void kernel_launch(void* const* d_in, const int* in_sizes, int n_in,
                              void* d_out, int out_size, void* d_ws, size_t ws_size,
                              hipStream_t stream)
{
    const float* vertices = (const float*)d_in[0];
    const float* mask     = (const float*)d_in[1];
    const float* query    = (const float*)d_in[2];
    const float* Wc       = (const float*)d_in[3];
    const float* bc       = (const float*)d_in[4];
    const float* wl       = (const float*)d_in[5];
    const float* bl       = (const float*)d_in[6];
    const float* Wr       = (const float*)d_in[7];
    const float* br       = (const float*)d_in[8];
    const float* temp     = (const float*)d_in[9];
    float* out = (float*)d_out;

    float* ws_heads = (float*)d_ws;                    // N * 256 floats
    float* ws_qh    = ws_heads + (size_t)N_ * NH_ * NE_; // 256 floats

    qh_kernel<<<1, 256, 0, stream>>>(Wc, bc, query, ws_qh);
    pool_kernel<<<N_, 256, 0, stream>>>(vertices, mask, ws_qh, wl, bl, temp, ws_heads);
    proj_kernel<<<8, 256, 0, stream>>>(ws_heads, Wr, br, out);
}